// PatchShift_55138790146356
// MI455X (gfx1250) — compile-verified
//
#include <hip/hip_runtime.h>

// PatchShift: out[b,t,w,h,i,j,:] = x[b, (t - smap[i,j]) & 7, w, h, i, j, :]
// x viewed as (b=8, t=8, w=64, h=4, i=7, j=7, c=32) f32.
// Pure permutation gather: 0 FLOPs, ~205.6 MB HBM traffic -> ~8.8 us floor @ 23.3 TB/s.
//
// Mapping: 1 block per (b,t,w) slab (4096 blocks), blockDim=392=49*8 so
// threadIdx spans (i,j,c4) directly (ij = tid>>3, no %49). t is block-uniform
// (SALU). The gather delta depends only on (i,j,t) -> computed ONCE per thread;
// the 4 items loop over h (stride 392 float4) reusing the same delta.
// Memory path: coalesced 128-bit non-temporal load/store (global_*_b128 th:TH_NT).

typedef float __attribute__((ext_vector_type(4))) f4;

__global__ __launch_bounds__(392) void patch_shift_kernel(const f4* __restrict__ in,
                                                          f4* __restrict__ out) {
    const int T_STRIDE4 = 64 * 4 * 49 * 8;  // t stride = 100352 float4

    // smap[i][j] = shifts3[i%3][j%3], shifts3 = {{-4,1,2},{-1,0,3},{-2,-3,4}}.
    // Nibble-packed (shift+4), index k = (i%3)*3 + (j%3), nibbles k=8..0: 8 1 2 7 4 3 6 5 0.
    const unsigned long long P = 0x812743650ULL;

    int tid = threadIdx.x;       // [0,392) covers (i,j,c4)
    int ij  = tid >> 3;          // i*7 + j   (once per thread)
    int i   = ij / 7;
    int j   = ij - i * 7;
    int k   = (i % 3) * 3 + (j % 3);
    int sh  = (int)((P >> (4 * k)) & 0xFULL) - 4;

    int t     = (blockIdx.x >> 6) & 7;              // blockIdx = (b*8+t)*64 + w (block-uniform)
    int delta = (((t - sh) & 7) - t) * T_STRIDE4;   // gather offset, invariant over h

    int base = blockIdx.x * (4 * 392) + tid;        // start of this (b,t,w) slab + lane
#pragma unroll
    for (int h = 0; h < 4; ++h) {
        int v = base + h * 392;                     // h stride = 49*8 float4
        f4 val = __builtin_nontemporal_load(&in[v + delta]);
        __builtin_nontemporal_store(val, &out[v]);
    }
}

extern "C" void kernel_launch(void* const* d_in, const int* in_sizes, int n_in,
                              void* d_out, int out_size, void* d_ws, size_t ws_size,
                              hipStream_t stream) {
    (void)n_in; (void)d_ws; (void)ws_size; (void)out_size;
    const f4* in = (const f4*)d_in[0];      // x: float32, 25,690,112 elements
    f4* out = (f4*)d_out;
    int elems_per_block = 4 * 49 * 8 * 4;   // h*i*j*c floats = 6272
    int blocks = in_sizes[0] / elems_per_block;  // 4096 = b*t*w
    patch_shift_kernel<<<blocks, 392, 0, stream>>>(in, out);
}